// DecoupledUncertaintyPrompt_50620484551132
// MI455X (gfx1250) — compile-verified
//
#include <hip/hip_runtime.h>
#include <hip/hip_bf16.h>
#include <stdint.h>

#define BB 2
#define CC 4
#define DD 128
#define HH 128
#define WW 128
#define DHW (DD*HH*WW)
#define K_TOP 20
#define RAD 10
#define CAP 4096

typedef unsigned int u32x4 __attribute__((ext_vector_type(4)));
typedef int          i32x8 __attribute__((ext_vector_type(8)));
typedef int          i32x4 __attribute__((ext_vector_type(4)));

__device__ __forceinline__ unsigned int umin_(unsigned int a, unsigned int b) { return a < b ? a : b; }
__device__ __forceinline__ unsigned int umax_(unsigned int a, unsigned int b) { return a > b ? a : b; }

// ---------------------------------------------------------------------------
// TDM: async DMA of a 2D tile (128 lines x 128 f32, line stride in elements)
// from global memory into LDS. D# per CDNA5 ISA ch.8 (groups 0/1; 2/3 zero).
// 6-arg builtin form (clang-23 / therock-10.0 headers).
// ---------------------------------------------------------------------------
__device__ __forceinline__ void tdm_load_tile_128x128(const float* src,
                                                      uint32_t lds_byte_off,
                                                      uint32_t line_stride_elems) {
    uint64_t ga = (uint64_t)(uintptr_t)src;
    uint32_t ga_lo = (uint32_t)__builtin_amdgcn_readfirstlane((int)(ga & 0xFFFFFFFFu));
    uint32_t ga_hi = (uint32_t)__builtin_amdgcn_readfirstlane((int)((ga >> 32) & 0x1FFFFFFu));
    uint32_t lds   = (uint32_t)__builtin_amdgcn_readfirstlane((int)lds_byte_off);
    uint32_t strd  = (uint32_t)__builtin_amdgcn_readfirstlane((int)line_stride_elems);

    u32x4 g0;
    g0[0] = 1u;                      // count=1 valid descriptor, gather off
    g0[1] = lds;                     // lds_addr (bytes)
    g0[2] = ga_lo;                   // global_addr[31:0]
    g0[3] = ga_hi | (2u << 30);      // global_addr[56:32] | type=2 ("image")

    i32x8 g1;
    g1[0] = (int)(2u << 16);         // wg_mask=0 | data_size=2 (4 bytes)
    g1[1] = (int)(128u << 16);       // atomic_barrier_addr=0 | tensor_dim0[15:0]=128
    g1[2] = (int)(128u << 16);       // tensor_dim0[31:16]=0 | tensor_dim1[15:0]=128
    g1[3] = (int)(128u << 16);       // tensor_dim1[31:16]=0 | tile_dim0=128
    g1[4] = 128;                     // tile_dim1=128 | tile_dim2=0
    g1[5] = (int)strd;               // tensor_dim0_stride[31:0] (elements)
    g1[6] = 0;                       // stride hi | tensor_dim1_stride lo
    g1[7] = 0;

    i32x4 gz4 = {0, 0, 0, 0};
    i32x8 gz8 = {0, 0, 0, 0, 0, 0, 0, 0};
    __builtin_amdgcn_tensor_load_to_lds(g0, g1, gz4, gz4, gz8, 0);
}

// ---------------------------------------------------------------------------
// K0: init header (uint-ordered min/max slots + candidate counters)
// ---------------------------------------------------------------------------
__global__ void init_kernel(unsigned int* hdr) {
    int t = threadIdx.x;
    if (t < 8)  hdr[t] = ((t & 1) == 0) ? 0x7F800000u : 0u; // even=min(+inf), odd=max(0)
    if (t >= 8 && t < 16) hdr[t] = 0u;                      // candidate counters
}

// ---------------------------------------------------------------------------
// K1: per-batch min/max of both entropy maps (values >= 0 -> uint ordering)
// ---------------------------------------------------------------------------
__global__ void minmax_kernel(const float* __restrict__ he,
                              const float* __restrict__ ha,
                              unsigned int* hdr) {
    __shared__ unsigned int s[4][256];
    int b = blockIdx.y;
    int tid = threadIdx.x;
    unsigned int hemin = 0x7F800000u, hemax = 0u, hamin = 0x7F800000u, hamax = 0u;
    size_t base = (size_t)b * DHW;
    for (int i = blockIdx.x * 256 + tid; i < DHW; i += gridDim.x * 256) {
        unsigned int e = __float_as_uint(he[base + i]);
        unsigned int a = __float_as_uint(ha[base + i]);
        hemin = umin_(hemin, e); hemax = umax_(hemax, e);
        hamin = umin_(hamin, a); hamax = umax_(hamax, a);
    }
    s[0][tid] = hemin; s[1][tid] = hemax; s[2][tid] = hamin; s[3][tid] = hamax;
    __syncthreads();
    for (int off = 128; off > 0; off >>= 1) {
        if (tid < off) {
            s[0][tid] = umin_(s[0][tid], s[0][tid + off]);
            s[1][tid] = umax_(s[1][tid], s[1][tid + off]);
            s[2][tid] = umin_(s[2][tid], s[2][tid + off]);
            s[3][tid] = umax_(s[3][tid], s[3][tid + off]);
        }
        __syncthreads();
    }
    if (tid == 0) {
        atomicMin(&hdr[b * 4 + 0], s[0][0]);
        atomicMax(&hdr[b * 4 + 1], s[1][0]);
        atomicMin(&hdr[b * 4 + 2], s[2][0]);
        atomicMax(&hdr[b * 4 + 3], s[3][0]);
    }
}

// ---------------------------------------------------------------------------
// K2: css = (1 - 0.5*(he_n + ha_n)) * seg_probs, float4 vectorized
// ---------------------------------------------------------------------------
__global__ void css_kernel(const float* __restrict__ he,
                           const float* __restrict__ ha,
                           const float* __restrict__ seg,
                           const unsigned int* __restrict__ hdr,
                           float* __restrict__ css) {
    int t = blockIdx.x * blockDim.x + threadIdx.x;   // vec4 index over B*DHW/4
    const int NV = DHW / 4;
    int b  = t / NV;
    int s4 = t - b * NV;
    float hemin = __uint_as_float(hdr[b * 4 + 0]);
    float hemax = __uint_as_float(hdr[b * 4 + 1]);
    float hamin = __uint_as_float(hdr[b * 4 + 2]);
    float hamax = __uint_as_float(hdr[b * 4 + 3]);
    float her = 1.0f / (hemax - hemin + 1e-8f);
    float har = 1.0f / (hamax - hamin + 1e-8f);

    float4 e = ((const float4*)he)[(size_t)b * NV + s4];
    float4 a = ((const float4*)ha)[(size_t)b * NV + s4];
    float4 sc;
    sc.x = 1.0f - 0.5f * ((e.x - hemin) * her + (a.x - hamin) * har);
    sc.y = 1.0f - 0.5f * ((e.y - hemin) * her + (a.y - hamin) * har);
    sc.z = 1.0f - 0.5f * ((e.z - hemin) * her + (a.z - hamin) * har);
    sc.w = 1.0f - 0.5f * ((e.w - hemin) * her + (a.w - hamin) * har);

    #pragma unroll
    for (int c = 0; c < CC; ++c) {
        size_t idx = (size_t)(b * CC + c) * NV + s4;
        float4 p = ((const float4*)seg)[idx];
        float4 o;
        o.x = sc.x * p.x; o.y = sc.y * p.y; o.z = sc.z * p.z; o.w = sc.w * p.w;
        ((float4*)css)[idx] = o;
    }
}

// ---------------------------------------------------------------------------
// K3/K4: 1D max-pool (window 21, -inf pad == edge clamp) along a strided line
// dimension. TDM stages the 128x128 tile into LDS (lane x -> bank x, conflict
// free). Sliding max via Gil-Werman chunks of 21: suffix-max array S (1 LDS
// read/entry) + running prefix max P (1 LDS read/output) -> ~2 LDS reads per
// output instead of 21. Fully unrolled so S[21] stays in VGPRs.
//   z-pass: lineStride=H*W, orthoStride=W   (css  -> tmp)
//   y-pass: lineStride=W,   orthoStride=H*W (tmp  -> tmp, in place; tile fully
//           staged in LDS before any write, tiles disjoint across blocks)
// ---------------------------------------------------------------------------
__global__ void pool_line_kernel(const float* __restrict__ in,
                                 float* __restrict__ out,
                                 int lineStride, int orthoStride) {
    __shared__ float tile[128 * 128];   // 64 KB
    int blk = blockIdx.x;               // bc*128 + o
    int o   = blk & 127;
    int bc  = blk >> 7;
    size_t base = (size_t)bc * DHW + (size_t)o * orthoStride;
    int tid = threadIdx.x;              // 0..127 == x

    if (tid < 32) {                     // wave 0 issues the DMA + waits on it
        uint32_t lds_off = (uint32_t)(uintptr_t)(&tile[0]);
        tdm_load_tile_128x128(in + base, lds_off, (uint32_t)lineStride);
        __builtin_amdgcn_s_wait_tensorcnt(0);
    }
    __syncthreads();

    #pragma unroll
    for (int c = 0; c < 7; ++c) {                 // chunks of 21 outputs
        const int t = c * 21;
        const int b = (t + 10 > 127) ? 127 : (t + 10);   // pivot
        const int e = (128 - t < 21) ? (128 - t) : 21;   // outputs in chunk

        // suffix max S[i] = max(tile[clamp(b-i)..b])
        float S[21];
        float run = -__builtin_inff();
        #pragma unroll
        for (int i = 0; i < 21; ++i) {
            int p = b - i; p = (p < 0) ? 0 : p;
            run = fmaxf(run, tile[p * 128 + tid]);
            S[i] = run;
        }

        // prefix max P = max(tile[b..covered]), extended one element per output
        float P = tile[b * 128 + tid];
        int covered = b;
        #pragma unroll
        for (int zz = 0; zz < 21; ++zz) {
            if (zz < e) {
                const int z  = t + zz;
                int up = z + RAD; if (up > 127) up = 127;
                if (up > covered) { P = fmaxf(P, tile[up * 128 + tid]); covered = up; }
                const int si = b - z + RAD;        // compile-time per (c,zz)
                float m = fmaxf(S[si], P);
                out[base + (size_t)z * lineStride + tid] = m;
            }
        }
    }
}

// ---------------------------------------------------------------------------
// K5: x-pass pool fused with NMS compare + candidate compaction
// (window scan along contiguous x: lanes read consecutive banks, conflict-free)
// ---------------------------------------------------------------------------
__global__ void xpass_cand_kernel(const float* __restrict__ tmp,
                                  const float* __restrict__ css,
                                  unsigned int* hdr,
                                  float* __restrict__ cand_score,
                                  int* __restrict__ cand_idx) {
    __shared__ float sp[8 * 128];
    __shared__ float sc[8 * 128];
    int blk  = blockIdx.x;              // ((bc*128 + z)*16 + yblk)
    int yblk = blk & 15;
    int z    = (blk >> 4) & 127;
    int bc   = blk >> 11;
    size_t rowbase = (size_t)bc * DHW + (size_t)z * (HH * WW) + (size_t)yblk * 8 * WW;
    int tid = threadIdx.x;              // 128 threads

    for (int i = tid; i < 1024; i += 128) {
        sp[i] = tmp[rowbase + i];
        sc[i] = css[rowbase + i];
    }
    __syncthreads();

    for (int r = 0; r < 8; ++r) {
        int x  = tid;
        int lo = x - RAD; if (lo < 0)   lo = 0;
        int hi = x + RAD; if (hi > 127) hi = 127;
        float m = -__builtin_inff();
        for (int k = lo; k <= hi; ++k) m = fmaxf(m, sp[r * 128 + k]);
        float v = sc[r * 128 + x];
        if (v == m) {                   // voxel is the 21^3 neighborhood max
            unsigned int pos = atomicAdd(&hdr[8 + bc], 1u);
            if (pos < CAP) {
                cand_score[bc * CAP + pos] = v;
                cand_idx  [bc * CAP + pos] = z * (HH * WW) + (yblk * 8 + r) * WW + x;
            }
        }
    }
}

// ---------------------------------------------------------------------------
// K6: top-20 selection per (b,c) from the compacted candidate list
// ---------------------------------------------------------------------------
__global__ void topk_kernel(const unsigned int* __restrict__ hdr,
                            float* __restrict__ cand_score,
                            const int* __restrict__ cand_idx,
                            float* __restrict__ out_coords,
                            void* out_labels, int labels_are_i64) {
    __shared__ float ss[256];
    __shared__ int   si[256];
    int bc = blockIdx.x;
    int b = bc / CC, c = bc % CC;
    int n = (int)umin_(hdr[8 + bc], (unsigned int)CAP);
    int tid = threadIdx.x;
    float* scores = cand_score + bc * CAP;
    const int* idxs = cand_idx + bc * CAP;

    for (int k = 0; k < K_TOP; ++k) {
        float best = -__builtin_inff(); int bslot = -1;
        for (int i = tid; i < n; i += 256) {
            float v = scores[i];
            if (v > best || (v == best && (unsigned)i < (unsigned)bslot)) { best = v; bslot = i; }
        }
        ss[tid] = best; si[tid] = bslot;
        __syncthreads();
        for (int off = 128; off > 0; off >>= 1) {
            if (tid < off) {
                float v2 = ss[tid + off]; int s2 = si[tid + off];
                if (v2 > ss[tid] || (v2 == ss[tid] && (unsigned)s2 < (unsigned)si[tid])) {
                    ss[tid] = v2; si[tid] = s2;
                }
            }
            __syncthreads();
        }
        int slot = si[0];
        if (tid == 0) {
            int vox = 0;
            if (slot >= 0) { vox = idxs[slot]; scores[slot] = -__builtin_inff(); }
            int d = vox / (HH * WW);
            int h = (vox % (HH * WW)) / WW;
            int w = vox % WW;
            size_t r = (size_t)(b * CC + c) * K_TOP + k;
            out_coords[r * 3 + 0] = (float)d;
            out_coords[r * 3 + 1] = (float)h;
            out_coords[r * 3 + 2] = (float)w;
            if (labels_are_i64) ((long long*)out_labels)[r] = (long long)c;
            else                ((float*)out_labels)[r]     = (float)c;
        }
        __syncthreads();
    }
}

// ---------------------------------------------------------------------------
extern "C" void kernel_launch(void* const* d_in, const int* in_sizes, int n_in,
                              void* d_out, int out_size, void* d_ws, size_t ws_size,
                              hipStream_t stream) {
    const float* he  = (const float*)d_in[0];
    const float* ha  = (const float*)d_in[1];
    const float* seg = (const float*)d_in[2];

    char* ws = (char*)d_ws;
    unsigned int* hdr  = (unsigned int*)ws;
    float* cand_score  = (float*)(ws + 256);
    int*   cand_idx    = (int*)(ws + 256 + (size_t)8 * CAP * 4);
    float* css         = (float*)(ws + (1u << 20));
    float* tmp         = (float*)(ws + (1u << 20) + (size_t)BB * CC * DHW * 4);

    float* out_coords  = (float*)d_out;
    const int NCOORD   = BB * CC * K_TOP * 3;   // 480 floats
    const int NLAB     = BB * CC * K_TOP;       // 160 labels
    int labels_are_i64 = (out_size >= NCOORD + 2 * NLAB) ? 1 : 0;
    void* out_labels   = (void*)(out_coords + NCOORD);

    init_kernel<<<1, 64, 0, stream>>>(hdr);

    dim3 gmm(256, BB);
    minmax_kernel<<<gmm, 256, 0, stream>>>(he, ha, hdr);

    int nvec = BB * DHW / 4;
    css_kernel<<<nvec / 256, 256, 0, stream>>>(he, ha, seg, hdr, css);

    // separable 21^3 max-pool
    pool_line_kernel<<<BB * CC * 128, 128, 0, stream>>>(css, tmp, HH * WW, WW); // z
    pool_line_kernel<<<BB * CC * 128, 128, 0, stream>>>(tmp, tmp, WW, HH * WW); // y (in place)
    xpass_cand_kernel<<<BB * CC * 128 * 16, 128, 0, stream>>>(tmp, css, hdr,
                                                              cand_score, cand_idx); // x + NMS

    topk_kernel<<<BB * CC, 256, 0, stream>>>(hdr, cand_score, cand_idx,
                                             out_coords, out_labels, labels_are_i64);
}